// MyModel_79869211836505
// MI455X (gfx1250) — compile-verified
//
#include <hip/hip_runtime.h>
#include <math.h>

typedef __attribute__((ext_vector_type(2))) float v2f;
typedef __attribute__((ext_vector_type(8))) float v8f;

#define DD 128
#define TOK 544              // tokens per batch per side (384 + 64 + 32 + 64)
#define BATCH 2
#define MROWS (BATCH*TOK)    // 1088 rows per half
#define MTOT (2*MROWS)       // 2176 stacked rows (target half ; ligase half)

__device__ __forceinline__ float lrelu(float x) {
    return x > 0.f ? x : 0.01f * x;
}

// ---------------------------------------------------------------------------
// Build A1 = [ v_t_d (1088x128) ; v_l_d (1088x128) ]
// v_d = concat(v_1[64], v_2[32], v_0[64]); v_t_d = [v_t(384); v_d], v_l_d = [v_l(384); v_d]
// ---------------------------------------------------------------------------
__global__ void concat_kernel(const float* __restrict__ v0,
                              const float* __restrict__ v1,
                              const float* __restrict__ v2,
                              const float* __restrict__ vt,
                              const float* __restrict__ vl,
                              float* __restrict__ A1) {
    const int r = blockIdx.x;        // 0..2175
    const int d = threadIdx.x;       // 0..127
    const int half = r / MROWS;      // 0 = target, 1 = ligase
    const int rr = r - half * MROWS;
    const int b  = rr / TOK;
    const int i  = rr - b * TOK;
    const float* src;
    if (i < 384) {
        src = (half == 0 ? vt : vl) + (b * 384 + i) * DD;
    } else {
        const int j = i - 384;
        if (j < 64)       src = v1 + (b * 64 + j) * DD;
        else if (j < 96)  src = v2 + (b * 32 + (j - 64)) * DD;
        else              src = v0 + (b * 64 + (j - 96)) * DD;
    }
    A1[r * DD + d] = src[d];
}

// ---------------------------------------------------------------------------
// C[m,n] = sum_k A[m,k] * W[n,k] + bias[n]   (i.e. A @ W.T + b), K = N = 128.
// Rows with row0 >= split use (W1b, bias1b), else (W0, bias0).
// act = 1 -> sigmoid.  One 16x16 tile per wave via V_WMMA_F32_16X16X4_F32.
// Block = 256 threads = 8 waves = 8 n-tiles; grid.x = M/16 m-tiles.
// ---------------------------------------------------------------------------
__global__ void gemm128_wmma(const float* __restrict__ A,
                             const float* __restrict__ W0, const float* __restrict__ bias0,
                             const float* __restrict__ W1b, const float* __restrict__ bias1b,
                             float* __restrict__ C, int split, int act) {
    const int lane = threadIdx.x & 31;
    const int wv   = threadIdx.x >> 5;          // n-tile 0..7
    const int row0 = blockIdx.x * 16;
    const int col0 = wv * 16;
    const float* W    = (row0 >= split) ? W1b   : W0;
    const float* bias = (row0 >= split) ? bias1b : bias0;

    // A 16x4 f32 layout: lanes 0-15 hold M=lane, K = {0,1}; lanes 16-31 hold K = {2,3}
    const int mr = lane & 15;
    const int kb = (lane >> 4) << 1;            // 0 or 2
    const float* ap = A + (row0 + mr) * DD + kb;
    // B 4x16: n = lane&15, same K split (B[k][n] = W[n][k] for the W.T product)
    const float* wp = W + (col0 + mr) * DD + kb;

    v8f acc = {0.f, 0.f, 0.f, 0.f, 0.f, 0.f, 0.f, 0.f};
    #pragma unroll
    for (int k0 = 0; k0 < DD; k0 += 4) {
        v2f a = *(const v2f*)(ap + k0);
        v2f b = *(const v2f*)(wp + k0);
        acc = __builtin_amdgcn_wmma_f32_16x16x4_f32(
                  /*neg_a=*/false, a, /*neg_b=*/false, b,
                  /*c_mod=*/(short)0, acc, /*reuse_a=*/false, /*reuse_b=*/false);
    }

    // C/D layout: VGPR r, lanes 0-15 -> M=r, N=lane; lanes 16-31 -> M=r+8, N=lane-16
    const int n = lane & 15;
    const int mbase = row0 + ((lane >> 4) << 3);
    const float bv = bias[col0 + n];
    #pragma unroll
    for (int r = 0; r < 8; ++r) {
        float v = acc[r] + bv;
        if (act) v = 1.0f / (1.0f + expf(-v));
        C[(mbase + r) * DD + col0 + n] = v;
    }
}

// ---------------------------------------------------------------------------
// Sbar[r, d] = (1/TOK) * sum_l lrelu(outerAtt[r, d] + innerAtt[b*TOK + l, d])
// 4 outer rows per block (inner stream reused 4x), 128 threads = one d each.
// ---------------------------------------------------------------------------
__global__ void pair_reduce(const float* __restrict__ outerAtt,
                            const float* __restrict__ innerAtt,
                            float* __restrict__ Sbar) {
    const int r0 = blockIdx.x * 4;   // 544 % 4 == 0 -> 4 rows stay in one batch
    const int b  = r0 / TOK;
    const int d  = threadIdx.x;
    const float x0 = outerAtt[(r0 + 0) * DD + d];
    const float x1 = outerAtt[(r0 + 1) * DD + d];
    const float x2 = outerAtt[(r0 + 2) * DD + d];
    const float x3 = outerAtt[(r0 + 3) * DD + d];
    float a0 = 0.f, a1 = 0.f, a2 = 0.f, a3 = 0.f;
    const float* ip = innerAtt + (b * TOK) * DD + d;
    #pragma unroll 4
    for (int l = 0; l < TOK; ++l) {
        const float y = ip[l * DD];
        float s;
        s = x0 + y; a0 += lrelu(s);
        s = x1 + y; a1 += lrelu(s);
        s = x2 + y; a2 += lrelu(s);
        s = x3 + y; a3 += lrelu(s);
    }
    const float inv = 1.0f / (float)TOK;
    Sbar[(r0 + 0) * DD + d] = a0 * inv;
    Sbar[(r0 + 1) * DD + d] = a1 * inv;
    Sbar[(r0 + 2) * DD + d] = a2 * inv;
    Sbar[(r0 + 3) * DD + d] = a3 * inv;
}

// ---------------------------------------------------------------------------
// x[b, half*128 + d] = sum_i V[b,i,d] * (0.5 + atte[b,i,d])
// grid = 4 blocks: blockIdx.x = b*2 + half; 128 threads (one d each).
// ---------------------------------------------------------------------------
__global__ void gate_sum(const float* __restrict__ A1,
                         const float* __restrict__ atte,   // 2176x128, [target;ligase]
                         float* __restrict__ x) {
    const int b    = blockIdx.x >> 1;
    const int half = blockIdx.x & 1;
    const int d    = threadIdx.x;
    const int base = (half * MROWS + b * TOK) * DD;
    float acc = 0.f;
    #pragma unroll 4
    for (int i = 0; i < TOK; ++i) {
        const float v = A1[base + i * DD + d];
        const float g = atte[base + i * DD + d];
        acc += v * (0.5f + g);
    }
    x[b * 256 + half * 128 + d] = acc;
}

// ---------------------------------------------------------------------------
// Fused 4-layer MLP for B=2 (pure GEMV territory, single block, LDS-staged).
// ---------------------------------------------------------------------------
__global__ __launch_bounds__(1024)
void mlp_kernel(const float* __restrict__ x,
                const float* __restrict__ W1, const float* __restrict__ b1,
                const float* __restrict__ W2, const float* __restrict__ b2,
                const float* __restrict__ W3, const float* __restrict__ b3,
                const float* __restrict__ Wo, const float* __restrict__ bo,
                float* __restrict__ out) {
    __shared__ float sx[2 * 256];
    __shared__ float f1[2 * 1024];
    __shared__ float f2[2 * 512];
    __shared__ float f3[2 * 256];
    const int t = threadIdx.x;
    if (t < 512) sx[t] = x[t];
    __syncthreads();
    {   // layer 1: 256 -> 1024
        const float* w = W1 + t * 256;
        for (int b = 0; b < 2; ++b) {
            float acc = b1[t];
            #pragma unroll 4
            for (int k = 0; k < 256; ++k) acc += sx[b * 256 + k] * w[k];
            f1[b * 1024 + t] = lrelu(acc);
        }
    }
    __syncthreads();
    if (t < 512) {   // layer 2: 1024 -> 512
        const float* w = W2 + t * 1024;
        for (int b = 0; b < 2; ++b) {
            float acc = b2[t];
            #pragma unroll 4
            for (int k = 0; k < 1024; ++k) acc += f1[b * 1024 + k] * w[k];
            f2[b * 512 + t] = lrelu(acc);
        }
    }
    __syncthreads();
    if (t < 256) {   // layer 3: 512 -> 256
        const float* w = W3 + t * 512;
        for (int b = 0; b < 2; ++b) {
            float acc = b3[t];
            #pragma unroll 4
            for (int k = 0; k < 512; ++k) acc += f2[b * 512 + k] * w[k];
            f3[b * 256 + t] = lrelu(acc);
        }
    }
    __syncthreads();
    if (t < 4) {     // output: 256 -> 2, predict laid out [b0o0, b0o1, b1o0, b1o1]
        const int b = t >> 1, o = t & 1;
        const float* w = Wo + o * 256;
        float acc = bo[o];
        for (int k = 0; k < 256; ++k) acc += f3[b * 256 + k] * w[k];
        out[t] = acc;
    }
}

// ---------------------------------------------------------------------------
extern "C" void kernel_launch(void* const* d_in, const int* in_sizes, int n_in,
                              void* d_out_v, int out_size, void* d_ws, size_t ws_size,
                              hipStream_t stream) {
    (void)in_sizes; (void)n_in; (void)out_size; (void)ws_size;
    const float* v0 = (const float*)d_in[0];
    const float* v1 = (const float*)d_in[1];
    const float* v2 = (const float*)d_in[2];
    const float* vt = (const float*)d_in[3];
    const float* vl = (const float*)d_in[4];
    const float* Wt = (const float*)d_in[5];
    const float* bt = (const float*)d_in[6];
    const float* Wl = (const float*)d_in[7];
    const float* bl = (const float*)d_in[8];
    const float* Wa = (const float*)d_in[9];
    const float* ba = (const float*)d_in[10];
    const float* W1 = (const float*)d_in[11];
    const float* b1 = (const float*)d_in[12];
    const float* W2 = (const float*)d_in[13];
    const float* b2 = (const float*)d_in[14];
    const float* W3 = (const float*)d_in[15];
    const float* b3 = (const float*)d_in[16];
    const float* Wo = (const float*)d_in[17];
    const float* bo = (const float*)d_in[18];
    float* out = (float*)d_out_v;

    // workspace layout (floats): A1 | att | Sbar | x  (~3.4 MB total)
    float* ws   = (float*)d_ws;
    float* A1   = ws;                       // 2176*128
    float* att  = A1   + MTOT * DD;         // 2176*128  [target_att ; ligase_att]
    float* Sbar = att  + MTOT * DD;         // 2176*128  [Sbar_t ; Sbar_l]
    float* xbuf = Sbar + MTOT * DD;         // 512

    // d_out = [predict(4) | target_atte(1088*128) | ligase_atte(1088*128)]
    float* atte = out + 4;                  // contiguous 2176x128 target

    // 1) concat token stacks
    concat_kernel<<<MTOT, DD, 0, stream>>>(v0, v1, v2, vt, vl, A1);
    // 2) att = [v_t_d @ Wt.T + bt ; v_l_d @ Wl.T + bl]  (row-split selects weights)
    gemm128_wmma<<<MTOT / 16, 256, 0, stream>>>(A1, Wt, bt, Wl, bl, att, MROWS, 0);
    // 3) mean-of-lrelu passes (mean commuted past @Wa.T)
    pair_reduce<<<MROWS / 4, DD, 0, stream>>>(att, att + MROWS * DD, Sbar);              // Sbar_t
    pair_reduce<<<MROWS / 4, DD, 0, stream>>>(att + MROWS * DD, att, Sbar + MROWS * DD); // Sbar_l
    // 4) atte = sigmoid(Sbar @ Wa.T + ba) -> straight into d_out
    gemm128_wmma<<<MTOT / 16, 256, 0, stream>>>(Sbar, Wa, ba, Wa, ba, atte, MTOT * 16, 1);
    // 5) gated token sums -> x (2 x 256)
    gate_sum<<<4, DD, 0, stream>>>(A1, atte, xbuf);
    // 6) fused MLP -> predict
    mlp_kernel<<<1, 1024, 0, stream>>>(xbuf, W1, b1, W2, b2, W3, b3, Wo, bo, out);
}